// BatchGridSubsampling_24670292148501
// MI455X (gfx1250) — compile-verified
//
#include <hip/hip_runtime.h>
#include <stdint.h>

// ---------------------------------------------------------------------------
// BatchGridSubsampling for MI455X (gfx1250).
//   * dense 16*11^3 voxel table (order == jnp.unique sorted key order)
//   * deterministic 2^-24 fixed-point int64 atomic accumulation
//   * TENSOR_LOAD_TO_LDS (TDM) staging of feats tiles + s_wait_tensorcnt
//   * v_wmma_f32_16x16x4_f32 for the final D = diag(1/count) x Sums scaling
// ---------------------------------------------------------------------------

typedef unsigned int v4u __attribute__((ext_vector_type(4)));
typedef int          v4i __attribute__((ext_vector_type(4)));
typedef int          v8i __attribute__((ext_vector_type(8)));
typedef float        v2f __attribute__((ext_vector_type(2)));
typedef float        v8f __attribute__((ext_vector_type(8)));

#define LSUB            11                   // local voxel coord in [0,10]
#define SLOTS_PER_BATCH (LSUB * LSUB * LSUB) // 1331
#define FIX_SCALE       16777216.0           // 2^24 fixed point

// ---------------------------------------------------------------- zero fill
__global__ void k_zero_bytes(uint8_t* __restrict__ p, size_t nbytes) {
  size_t stride = (size_t)gridDim.x * blockDim.x;
  size_t i = (size_t)blockIdx.x * blockDim.x + threadIdx.x;
  size_t n16 = nbytes >> 4;
  uint4 z = make_uint4(0u, 0u, 0u, 0u);
  for (size_t k = i; k < n16; k += stride) ((uint4*)p)[k] = z;
  if (i == 0) {
    for (size_t t = n16 << 4; t < nbytes; ++t) p[t] = 0;
  }
}

// ------------------------------------------------------- batch_len prefix
__global__ void k_batch_prefix(const int* __restrict__ blen, int B,
                               int* __restrict__ bpre) {
  if (threadIdx.x == 0 && blockIdx.x == 0) {
    int acc = 0;
    for (int b = 0; b < B; ++b) { bpre[b] = acc; acc += blen[b]; }
    bpre[B] = acc;
  }
}

// ---------------------------------------------------------- accumulation
__global__ __launch_bounds__(256) void k_accum(
    const float* __restrict__ pts, const float* __restrict__ fts,
    const int* __restrict__ bpre, int n, int B, int D,
    long long* __restrict__ sums, unsigned* __restrict__ cnt) {
  __shared__ float tile[256 * 33];  // sole LDS object -> group-segment offset 0
  const int blockStart = blockIdx.x * 256;
  const int rem = min(256, n - blockStart);
  const bool useTdm = (D == 32);

  if (useTdm && threadIdx.x < 32) {  // wave 0 only: TDM is a per-wave op
    unsigned elems = (unsigned)rem * 32u;
    unsigned long long ga =
        (unsigned long long)(uintptr_t)(fts + (size_t)blockStart * 32);
    // D# group 0: count=1, lds_addr=0 (tile at LDS offset 0), global addr,
    // type=2 ("image") in bits[127:126].
    v4u g0;
    g0[0] = 1u;
    g0[1] = 0u;
    g0[2] = (unsigned)(ga & 0xffffffffull);
    g0[3] = (unsigned)((ga >> 32) & 0x01ffffffull) | (2u << 30);
    // D# group 1: data_size=2 (4B), pad_enable, pad_interval=4 (every 32
    // dwords), pad_amount=0 (1 dword) -> LDS row stride 33 (conflict-free);
    // tensor_dim0 = tile_dim0 = elems, tensor_dim1 = tile_dim1 = 1.
    v8i g1;
    g1[0] = (int)((2u << 16) | (1u << 20) | (4u << 22));
    g1[1] = (int)((elems & 0xffffu) << 16);                 // tensor_dim0 lo
    g1[2] = (int)(((elems >> 16) & 0xffffu) | (1u << 16));  // dim0 hi, dim1=1
    g1[3] = (int)((elems & 0xffffu) << 16);                 // tile_dim0
    g1[4] = 1;                                              // tile_dim1 = 1
    g1[5] = (int)elems;                                     // dim0 stride lo
    g1[6] = 0;
    g1[7] = 0;
    v4i gz = (v4i){0, 0, 0, 0};
#if defined(__clang_major__) && (__clang_major__ >= 23)
    v8i gz8 = (v8i){0, 0, 0, 0, 0, 0, 0, 0};
    __builtin_amdgcn_tensor_load_to_lds(g0, g1, gz, gz, gz8, 0);
#else
    __builtin_amdgcn_tensor_load_to_lds(g0, g1, gz, gz, 0);
#endif
    __builtin_amdgcn_s_wait_tensorcnt(0);
  }
  __syncthreads();

  const int p = blockStart + (int)threadIdx.x;
  if (p >= n) return;
  if (p + 256 < n) __builtin_prefetch(&pts[(size_t)(p + 256) * 3], 0, 1);

  int b = 0;
  while (b + 1 < B && p >= bpre[b + 1]) ++b;
  const float off = 5.0f * (float)b;  // OFFSET = 5, exact in f32

  // Bit-for-bit the reference math: f32 add, f32 divide by 0.1f, floor.
  float x = pts[(size_t)p * 3 + 0] + off;
  float y = pts[(size_t)p * 3 + 1] + off;
  float z = pts[(size_t)p * 3 + 2] + off;
  int lx = (int)floorf(x / 0.1f) - 50 * b;
  int ly = (int)floorf(y / 0.1f) - 50 * b;
  int lz = (int)floorf(z / 0.1f) - 50 * b;
  lx = max(0, min(LSUB - 1, lx));
  ly = max(0, min(LSUB - 1, ly));
  lz = max(0, min(LSUB - 1, lz));
  const int slot = ((b * LSUB + lx) * LSUB + ly) * LSUB + lz;

  long long* s = sums + (size_t)slot * (size_t)(3 + D);
  atomicAdd(&cnt[slot], 1u);
  atomicAdd((unsigned long long*)&s[0],
            (unsigned long long)__double2ll_rn((double)x * FIX_SCALE));
  atomicAdd((unsigned long long*)&s[1],
            (unsigned long long)__double2ll_rn((double)y * FIX_SCALE));
  atomicAdd((unsigned long long*)&s[2],
            (unsigned long long)__double2ll_rn((double)z * FIX_SCALE));
  const float* frow = fts + (size_t)p * (size_t)D;
  for (int j = 0; j < D; ++j) {
    float f = useTdm ? tile[threadIdx.x * 33 + j] : frow[j];
    atomicAdd((unsigned long long*)&s[3 + j],
              (unsigned long long)__double2ll_rn((double)f * FIX_SCALE));
  }
}

// --------------------------------------------- occupancy scan + pool_batch
__global__ __launch_bounds__(1024) void k_scan(const unsigned* __restrict__ cnt,
                                               unsigned* __restrict__ rank,
                                               float* __restrict__ pool_batch,
                                               int B, int nslot) {
  __shared__ unsigned s[1024];
  __shared__ unsigned carry;
  if (threadIdx.x == 0) carry = 0u;
  __syncthreads();
  for (int base = 0; base < nslot; base += 1024) {
    int i = base + (int)threadIdx.x;
    unsigned f = (i < nslot && cnt[i] > 0u) ? 1u : 0u;
    s[threadIdx.x] = f;
    __syncthreads();
    for (int off = 1; off < 1024; off <<= 1) {
      unsigned v = (threadIdx.x >= (unsigned)off) ? s[threadIdx.x - off] : 0u;
      __syncthreads();
      s[threadIdx.x] += v;
      __syncthreads();
    }
    if (i < nslot) rank[i] = carry + s[threadIdx.x] - f;  // exclusive rank
    __syncthreads();
    if (threadIdx.x == 0) carry += s[1023];
    __syncthreads();
  }
  if ((int)threadIdx.x < B) {
    int b = (int)threadIdx.x;
    unsigned start = rank[b * SLOTS_PER_BATCH];
    unsigned end = (b == B - 1) ? carry : rank[(b + 1) * SLOTS_PER_BATCH];
    pool_batch[b] = (float)(end - start);
  }
}

// --------------------------- pool_feats via WMMA: D = diag(1/count) x Sums
// One wave per 16 consecutive slots. A = diagonal scale block (4 K-chunks),
// B = 2^-24-descaled sums tile (16 slots x 16 cols), C accumulates chunks.
// Off-diagonal A entries are exact zeros, so each output element is
// inv_i * S[i,j] with a single f32 rounding — numerically equal to scalar.
__global__ __launch_bounds__(256) void k_write_feats_wmma(
    const long long* __restrict__ sums, const unsigned* __restrict__ cnt,
    const unsigned* __restrict__ rank, float* __restrict__ out_fts,
    int ngroup) {
  const int lane = (int)(threadIdx.x & 31u);
  const int wave = (int)(threadIdx.x >> 5);
  const int group = blockIdx.x * 8 + wave;  // 16 slots per group
  if (group >= ngroup) return;              // wave-uniform: EXEC stays all-1s
  const int slotBase = group * 16;
  const int half = lane >> 4;               // 0: lanes 0-15, 1: lanes 16-31
  const int nIdx = lane & 15;

  // A-matrix 16x4 layout: lanes 0-15 hold M=lane, K={4k,4k+1}; lanes 16-31
  // hold M=lane-16, K={4k+2,4k+3}. This lane's A row:
  const int mA = lane & 15;
  const unsigned cA = cnt[slotBase + mA];
  const float invA = (cA > 0u) ? (1.0f / (float)cA) : 0.0f;

  const double toF = 1.0 / FIX_SCALE;
  const size_t stride = 35;  // 3 + 32
  const long long* base = sums + (size_t)slotBase * stride + 3;

  for (int colChunk = 0; colChunk < 2; ++colChunk) {
    const int colBase = colChunk * 16;
    v8f c = {};
    for (int k = 0; k < 4; ++k) {
      const int kk = 4 * k + 2 * half;  // this lane's two K indices
      v2f a;
      a.x = (mA == kk)     ? invA : 0.0f;
      a.y = (mA == kk + 1) ? invA : 0.0f;
      // B-matrix 4x16: lanes 0-15 hold N=lane, K={4k,4k+1}; lanes 16-31
      // hold N=lane-16, K={4k+2,4k+3}.  B[k_row, n] = descaled S[k_row, n].
      v2f b;
      b.x = (float)((double)base[(size_t)kk * stride + colBase + nIdx] * toF);
      b.y = (float)((double)base[(size_t)(kk + 1) * stride + colBase + nIdx] *
                    toF);
      c = __builtin_amdgcn_wmma_f32_16x16x4_f32(false, a, false, b, (short)0,
                                                c, false, false);
    }
    // D layout: VGPR v holds row M = v + 8*half, col N = nIdx.
    for (int v = 0; v < 8; ++v) {
      const int slot = slotBase + v + 8 * half;
      if (cnt[slot] > 0u)
        out_fts[(size_t)rank[slot] * 32 + colBase + nIdx] = c[v];
    }
  }
}

// --------------------------- scalar writeout: points (+ feats fallback/tail)
__global__ void k_write_scalar(const long long* __restrict__ sums,
                               const unsigned* __restrict__ cnt,
                               const unsigned* __restrict__ rank, int D,
                               int featsFrom, float* __restrict__ out_pts,
                               float* __restrict__ out_fts, int nslot) {
  int sIdx = blockIdx.x * blockDim.x + threadIdx.x;
  if (sIdx >= nslot) return;
  unsigned c = cnt[sIdx];
  if (c == 0u) return;
  unsigned r = rank[sIdx];
  const long long* q = sums + (size_t)sIdx * (size_t)(3 + D);
  double inv = 1.0 / ((double)c * FIX_SCALE);
  out_pts[(size_t)r * 3 + 0] = (float)((double)q[0] * inv);
  out_pts[(size_t)r * 3 + 1] = (float)((double)q[1] * inv);
  out_pts[(size_t)r * 3 + 2] = (float)((double)q[2] * inv);
  if (sIdx >= featsFrom) {  // WMMA kernel covers slots < featsFrom
    for (int j = 0; j < D; ++j)
      out_fts[(size_t)r * D + j] = (float)((double)q[3 + j] * inv);
  }
}

// ---------------------------------------------------------------- launcher
extern "C" void kernel_launch(void* const* d_in, const int* in_sizes, int n_in,
                              void* d_out, int out_size, void* d_ws,
                              size_t ws_size, hipStream_t stream) {
  const float* pts = (const float*)d_in[0];
  const float* fts = (const float*)d_in[1];
  const int* blen = (const int*)d_in[2];
  const int n = in_sizes[0] / 3;
  const int B = in_sizes[2];
  const int D = (n > 0) ? in_sizes[1] / n : 0;
  const int nslot = B * SLOTS_PER_BATCH;

  float* out = (float*)d_out;
  float* out_pts = out;
  float* out_fts = out + (size_t)n * 3;
  float* out_bat = out + (size_t)n * (size_t)(3 + D);

  // ws: [int64 sums nslot*(3+D)] [u32 cnt nslot] [u32 rank nslot] [int bpre]
  long long* sums = (long long*)d_ws;
  size_t sumsBytes = (size_t)nslot * (size_t)(3 + D) * sizeof(long long);
  unsigned* cnt = (unsigned*)((char*)d_ws + sumsBytes);
  unsigned* rank = cnt + nslot;
  int* bpre = (int*)(rank + nslot);

  k_zero_bytes<<<2048, 256, 0, stream>>>((uint8_t*)d_out,
                                         (size_t)out_size * sizeof(float));
  k_zero_bytes<<<512, 256, 0, stream>>>(
      (uint8_t*)d_ws, sumsBytes + (size_t)nslot * sizeof(unsigned));
  k_batch_prefix<<<1, 1, 0, stream>>>(blen, B, bpre);
  k_accum<<<(n + 255) / 256, 256, 0, stream>>>(pts, fts, bpre, n, B, D, sums,
                                               cnt);
  k_scan<<<1, 1024, 0, stream>>>(cnt, rank, out_bat, B, nslot);

  const int ngroup = (D == 32) ? (nslot / 16) : 0;  // full 16-slot groups
  const int featsFrom = ngroup * 16;
  if (ngroup > 0)
    k_write_feats_wmma<<<(ngroup + 7) / 8, 256, 0, stream>>>(sums, cnt, rank,
                                                             out_fts, ngroup);
  k_write_scalar<<<(nslot + 255) / 256, 256, 0, stream>>>(
      sums, cnt, rank, D, featsFrom, out_pts, out_fts, nslot);
}